// GAT_12781822673317
// MI455X (gfx1250) — compile-verified
//
#include <hip/hip_runtime.h>
#include <hip/hip_bf16.h>

#define N_NODES   100000
#define N_EDGES   1600000
#define E_TOT     1700000      // edges + self loops
#define IN_C      128
#define HIDC      32
#define HEADS     4
#define OUT_C     40
#define N_GRAPHS  64
#define NEG_SLOPE 0.2f

typedef float v2f __attribute__((ext_vector_type(2)));
typedef float v8f __attribute__((ext_vector_type(8)));

// ---------------- utility fills ----------------
__global__ void fill_f32(float* p, float v, int n) {
    int i = blockIdx.x * blockDim.x + threadIdx.x;
    if (i < n) p[i] = v;
}
__global__ void fill_u32(unsigned* p, unsigned v, int n) {
    int i = blockIdx.x * blockDim.x + threadIdx.x;
    if (i < n) p[i] = v;
}

// Transpose (and zero-pad) weights: BT[c*128 + k] = (c < NC) ? W[k*NC + c] : 0
__global__ void transpose_w(const float* __restrict__ W, float* __restrict__ BT,
                            int NC, int NCPAD) {
    int i = blockIdx.x * blockDim.x + threadIdx.x;
    if (i >= 128 * NCPAD) return;
    int c = i >> 7, k = i & 127;            // i = c*128 + k
    BT[i] = (c < NC) ? W[k * NC + c] : 0.f;
}

// ---------------- WMMA f32 GEMM: D[N x NCOUT] = A[N x 128] * B[128 x NCOUT] ----------------
// BT is the column-major (transposed, padded to NCPAD cols) weight: BT[col*128 + k].
// One block = 16 rows of A staged in LDS; wave w owns column tile [16w, 16w+16).
// blockDim.x must be (NCPAD/16)*32. Inner loop: ds_load_b64 + global_load_b64 + wmma.
template <int NCPAD, int NCOUT>
__global__ void gemm16_wmma(const float* __restrict__ A, const float* __restrict__ BT,
                            float* __restrict__ D) {
    __shared__ float As[16 * 132];          // pad 128->132: conflict-free A-fragment reads
    const int rowbase = blockIdx.x * 16;
    for (int i = threadIdx.x; i < 16 * 128; i += blockDim.x) {
        int r = i >> 7, c = i & 127;
        As[r * 132 + c] = A[(size_t)(rowbase + r) * 128 + c];
    }
    __syncthreads();

    const int wave = threadIdx.x >> 5;
    const int lane = threadIdx.x & 31;
    const int hh   = lane >> 4;             // half-wave: K-pair select / M-half on output
    const int lrow = lane & 15;
    const int col  = wave * 16 + lrow;      // always < NCPAD

    const float* ap = As + lrow * 132 + 2 * hh;            // 8B-aligned (all terms even)
    const float* bp = BT + (size_t)col * 128 + 2 * hh;     // contiguous K pair per lane

    v8f acc = {0.f, 0.f, 0.f, 0.f, 0.f, 0.f, 0.f, 0.f};
#pragma unroll
    for (int k = 0; k < 128; k += 4) {
        v2f a = *(const v2f*)(ap + k);      // A: K=(k+2hh), K=(k+2hh+1), row=lrow
        v2f b = *(const v2f*)(bp + k);      // B: same K pair, fixed column
        acc = __builtin_amdgcn_wmma_f32_16x16x4_f32(false, a, false, b,
                                                    (short)0, acc, false, false);
    }
    if (NCPAD == NCOUT || col < NCOUT) {
#pragma unroll
        for (int r = 0; r < 8; ++r)
            D[(size_t)(rowbase + hh * 8 + r) * NCOUT + col] = acc[r];
    }
}

// ---------------- per-node attention coefficients ----------------
__global__ void node_alpha(const float* __restrict__ H, const float* __restrict__ av_s,
                           const float* __restrict__ av_d, float* __restrict__ outs,
                           float* __restrict__ outd, int heads, int cper) {
    int tid = blockIdx.x * blockDim.x + threadIdx.x;
    if (tid >= N_NODES * heads) return;
    int n = tid / heads, h = tid % heads;
    const float* row = H + (size_t)n * heads * cper + h * cper;
    float s = 0.f, d = 0.f;
    for (int c = 0; c < cper; ++c) {
        float v = row[c];
        s += v * av_s[h * cper + c];
        d += v * av_d[h * cper + c];
    }
    outs[tid] = s;
    outd[tid] = d;
}

// ---------------- edge helpers ----------------
__device__ __forceinline__ void edge_sd(const int* __restrict__ ei, int e, int& s, int& d) {
    if (e < N_EDGES) { s = ei[e]; d = ei[N_EDGES + e]; }
    else             { s = e - N_EDGES; d = s; }      // appended self-loops
}

// monotonic float-max via int atomicMax / uint atomicMin (init = 0xFF800000 = -inf)
__device__ __forceinline__ void atomic_max_f32(unsigned* addr, float v) {
    int iv = __float_as_int(v);
    if (iv >= 0) atomicMax((int*)addr, iv);
    else         atomicMin(addr, (unsigned)iv);
}

template <int H>
__global__ void edge_logits(const int* __restrict__ ei, const float* __restrict__ as_,
                            const float* __restrict__ ad_, float* __restrict__ ex,
                            unsigned* __restrict__ mbits) {
    int tid = blockIdx.x * blockDim.x + threadIdx.x;
    if (tid >= E_TOT * H) return;
    int e = tid / H, h = tid % H;
    int s, d; edge_sd(ei, e, s, d);
    float v = as_[s * H + h] + ad_[d * H + h];
    v = (v > 0.f) ? v : NEG_SLOPE * v;                // LeakyReLU(0.2)
    ex[tid] = v;
    atomic_max_f32(&mbits[d * H + h], v);
}

template <int H>
__global__ void edge_exp(const int* __restrict__ ei, float* __restrict__ ex,
                         const unsigned* __restrict__ mbits, float* __restrict__ denom) {
    int tid = blockIdx.x * blockDim.x + threadIdx.x;
    if (tid >= E_TOT * H) return;
    int e = tid / H, h = tid % H;
    int s, d; edge_sd(ei, e, s, d);
    float m = __int_as_float((int)mbits[d * H + h]);
    float ev = expf(ex[tid] - m);
    ex[tid] = ev;
    atomicAdd(&denom[d * H + h], ev);
}

// one thread per (edge, channel): coalesced gather of h[src] row, per-lane-distinct atomic scatter
template <int CH, int CperH>
__global__ void edge_aggr(const int* __restrict__ ei, const float* __restrict__ Hin,
                          const float* __restrict__ ex, const float* __restrict__ denom,
                          float* __restrict__ out) {
    int tid = blockIdx.x * blockDim.x + threadIdx.x;
    if (tid >= E_TOT * CH) return;
    const int Hh = CH / CperH;
    int e = tid / CH, c = tid % CH, h = c / CperH;
    int s, d; edge_sd(ei, e, s, d);
    float alpha = ex[e * Hh + h] / (denom[d * Hh + h] + 1e-16f);
    atomicAdd(&out[(size_t)d * CH + c], Hin[(size_t)s * CH + c] * alpha);
}

__global__ void elu_bias(const float* __restrict__ in, const float* __restrict__ b,
                         float* __restrict__ out) {
    int i = blockIdx.x * blockDim.x + threadIdx.x;
    if (i >= N_NODES * IN_C) return;
    float v = in[i] + b[i & 127];
    out[i] = (v > 0.f) ? v : (expf(v) - 1.f);         // ELU(alpha=1)
}

// ---------------- graph pooling ----------------
__global__ void pool_count(const int* __restrict__ batch, float* __restrict__ counts) {
    int n = blockIdx.x * blockDim.x + threadIdx.x;
    if (n >= N_NODES) return;
    atomicAdd(&counts[batch[n]], 1.f);
}
__global__ void pool_sum(const float* __restrict__ out2, const float* __restrict__ b2,
                         const int* __restrict__ batch, float* __restrict__ sums) {
    int i = blockIdx.x * blockDim.x + threadIdx.x;
    if (i >= N_NODES * OUT_C) return;
    int n = i / OUT_C, c = i % OUT_C;
    atomicAdd(&sums[batch[n] * OUT_C + c], out2[i] + b2[c]);
}
__global__ void pool_final(const float* __restrict__ sums, const float* __restrict__ counts,
                           float* __restrict__ out) {
    int i = blockIdx.x * blockDim.x + threadIdx.x;
    if (i >= N_GRAPHS * OUT_C) return;
    out[i] = sums[i] / fmaxf(counts[i / OUT_C], 1.f);
}

// ---------------- launch ----------------
extern "C" void kernel_launch(void* const* d_in, const int* in_sizes, int n_in,
                              void* d_out, int out_size, void* d_ws, size_t ws_size,
                              hipStream_t stream) {
    const float* x   = (const float*)d_in[0];
    const int*   ei  = (const int*)d_in[1];     // [2, N_EDGES]
    const int*   bat = (const int*)d_in[2];
    const float* W1  = (const float*)d_in[3];   // [128,128]
    const float* a1s = (const float*)d_in[4];
    const float* a1d = (const float*)d_in[5];
    const float* b1  = (const float*)d_in[6];
    const float* W2  = (const float*)d_in[7];   // [128,40]
    const float* a2s = (const float*)d_in[8];
    const float* a2d = (const float*)d_in[9];
    const float* b2  = (const float*)d_in[10];

    char* w = (char*)d_ws;
    size_t off = 0;
    auto alloc = [&](size_t bytes) -> char* {
        char* p = w + off;
        off += (bytes + 255) & ~(size_t)255;
        return p;
    };
    float*    h1   = (float*)   alloc((size_t)N_NODES * 128 * 4);
    float*    out1 = (float*)   alloc((size_t)N_NODES * 128 * 4);
    float*    ex1  = (float*)   alloc((size_t)E_TOT * HEADS * 4);
    float*    as1  = (float*)   alloc((size_t)N_NODES * HEADS * 4);
    float*    ad1  = (float*)   alloc((size_t)N_NODES * HEADS * 4);
    unsigned* m1   = (unsigned*)alloc((size_t)N_NODES * HEADS * 4);
    float*    dn1  = (float*)   alloc((size_t)N_NODES * HEADS * 4);
    float*    h2   = (float*)   alloc((size_t)N_NODES * OUT_C * 4);
    float*    out2 = (float*)   alloc((size_t)N_NODES * OUT_C * 4);
    float*    as2  = (float*)   alloc((size_t)N_NODES * 4);
    float*    ad2  = (float*)   alloc((size_t)N_NODES * 4);
    unsigned* m2   = (unsigned*)alloc((size_t)N_NODES * 4);
    float*    dn2  = (float*)   alloc((size_t)N_NODES * 4);
    float*    ex2  = (float*)   alloc((size_t)E_TOT * 4);
    float*    sums = (float*)   alloc((size_t)N_GRAPHS * OUT_C * 4);
    float*    cnts = (float*)   alloc((size_t)N_GRAPHS * 4);
    float*    BT1  = (float*)   alloc((size_t)128 * 128 * 4);   // W1^T
    float*    BT2  = (float*)   alloc((size_t)48 * 128 * 4);    // W2^T zero-padded to 48 cols

    auto g = [](long n) { return (int)((n + 255) / 256); };

    // ---- init accumulators + weight transposes (every call: graph-replay safe) ----
    fill_f32<<<g(N_NODES * 128), 256, 0, stream>>>(out1, 0.f, N_NODES * 128);
    fill_u32<<<g(N_NODES * HEADS), 256, 0, stream>>>(m1, 0xFF800000u, N_NODES * HEADS);
    fill_f32<<<g(N_NODES * HEADS), 256, 0, stream>>>(dn1, 0.f, N_NODES * HEADS);
    fill_f32<<<g(N_NODES * OUT_C), 256, 0, stream>>>(out2, 0.f, N_NODES * OUT_C);
    fill_u32<<<g(N_NODES), 256, 0, stream>>>(m2, 0xFF800000u, N_NODES);
    fill_f32<<<g(N_NODES), 256, 0, stream>>>(dn2, 0.f, N_NODES);
    fill_f32<<<g(N_GRAPHS * OUT_C), 256, 0, stream>>>(sums, 0.f, N_GRAPHS * OUT_C);
    fill_f32<<<g(N_GRAPHS), 256, 0, stream>>>(cnts, 0.f, N_GRAPHS);
    transpose_w<<<g(128 * 128), 256, 0, stream>>>(W1, BT1, 128, 128);
    transpose_w<<<g(128 * 48), 256, 0, stream>>>(W2, BT2, OUT_C, 48);

    // ---- layer 1: GAT(128 -> 4x32, concat) ----
    gemm16_wmma<128, 128><<<N_NODES / 16, 256, 0, stream>>>(x, BT1, h1);
    node_alpha<<<g(N_NODES * HEADS), 256, 0, stream>>>(h1, a1s, a1d, as1, ad1, HEADS, HIDC);
    edge_logits<HEADS><<<g((long)E_TOT * HEADS), 256, 0, stream>>>(ei, as1, ad1, ex1, m1);
    edge_exp<HEADS><<<g((long)E_TOT * HEADS), 256, 0, stream>>>(ei, ex1, m1, dn1);
    edge_aggr<128, HIDC><<<g((long)E_TOT * 128), 256, 0, stream>>>(ei, h1, ex1, dn1, out1);
    elu_bias<<<g(N_NODES * 128), 256, 0, stream>>>(out1, b1, h1);   // h1 := ELU(out1 + b1)

    // ---- layer 2: GAT(128 -> 40, 1 head) ----
    gemm16_wmma<48, OUT_C><<<N_NODES / 16, 96, 0, stream>>>(h1, BT2, h2);
    node_alpha<<<g(N_NODES), 256, 0, stream>>>(h2, a2s, a2d, as2, ad2, 1, OUT_C);
    edge_logits<1><<<g(E_TOT), 256, 0, stream>>>(ei, as2, ad2, ex2, m2);
    edge_exp<1><<<g(E_TOT), 256, 0, stream>>>(ei, ex2, m2, dn2);
    edge_aggr<OUT_C, OUT_C><<<g((long)E_TOT * OUT_C), 256, 0, stream>>>(ei, h2, ex2, dn2, out2);

    // ---- global mean pool ----
    pool_count<<<g(N_NODES), 256, 0, stream>>>(bat, cnts);
    pool_sum<<<g((long)N_NODES * OUT_C), 256, 0, stream>>>(out2, b2, bat, sums);
    pool_final<<<g(N_GRAPHS * OUT_C), 256, 0, stream>>>(sums, cnts, (float*)d_out);
}